// Geodesic_integrator_43688407335619
// MI455X (gfx1250) — compile-verified
//
#include <hip/hip_runtime.h>
#include <math.h>

#define HH 1024
#define WW 1024
#define RAD 18
#define KS 37
#define NPIX (HH * WW)
#define INV_N 0.0625f /* 1/16 */

// padded banded-weight table: index d+15, d = p - M in [-15, 51]
#define WTAB 67

typedef __attribute__((ext_vector_type(2))) float v2f;
typedef __attribute__((ext_vector_type(8))) float v8f;

// ---------------------------------------------------------------------------
// force: f = -z * grad(img), central differences, replicate edges
// ---------------------------------------------------------------------------
__global__ __launch_bounds__(256) void force_k(const float* __restrict__ img,
                                               const float* __restrict__ z,
                                               float* __restrict__ fx,
                                               float* __restrict__ fy) {
  int idx = blockIdx.x * 256 + threadIdx.x;
  int y = idx >> 10;
  int x = idx & (WW - 1);
  int xm = max(x - 1, 0), xp = min(x + 1, WW - 1);
  int ym = max(y - 1, 0), yp = min(y + 1, HH - 1);
  float gx = 0.5f * (img[y * WW + xp] - img[y * WW + xm]);
  float gy = 0.5f * (img[yp * WW + x] - img[ym * WW + x]);
  float zc = z[idx];
  fx[idx] = -zc * gx;
  fy[idx] = -zc * gy;
}

// ---------------------------------------------------------------------------
// Vertical 37-tap Gaussian blur, output TRANSPOSED. One wave computes a
// 32x16 output tile as two 16x16 WMMA accumulations (c0: rows y0..y0+15,
// c1: rows y0+16..y0+31) that SHARE the B-operand registers:
//   c0 += A[k] x B[k]   (k = 0..12)
//   c1 += A[k] x B[k+4]
// A = banded Toeplitz weights, read branchlessly from a zero-padded LDS
// table and hoisted into registers. All 17 B chunks (68 input rows, row-
// contiguous 64B segments) are preloaded, then WMMAs run back-to-back.
// 16x16 LDS transpose (stride 17) -> coalesced transposed stores.
// 4 waves/block; blockIdx.y selects channel.
// ---------------------------------------------------------------------------
__global__ __launch_bounds__(128) void blurV_T(const float* __restrict__ in0,
                                               const float* __restrict__ in1,
                                               float* __restrict__ out0T,
                                               float* __restrict__ out1T) {
  __shared__ float w[KS];
  __shared__ float winv;
  __shared__ float wtab[WTAB];
  __shared__ float tile[4][16 * 17];

  int tid = threadIdx.x;
  if (tid < KS) {
    float t = (float)(tid - RAD) * (1.0f / 6.0f);
    w[tid] = expf(-0.5f * t * t);
  }
  __syncthreads();
  if (tid == 0) {
    float s = 0.0f;
    for (int i = 0; i < KS; ++i) s += w[i];
    winv = 1.0f / s;
  }
  __syncthreads();
  if (tid < WTAB) {
    int d = tid - 15;  // p - M
    wtab[tid] = (d >= 0 && d < KS) ? w[d] * winv : 0.0f;
  }
  __syncthreads();

  const float* __restrict__ in = blockIdx.y ? in1 : in0;
  float* __restrict__ outT = blockIdx.y ? out1T : out0T;

  int wave = tid >> 5;
  int lane = tid & 31;
  int half = lane >> 4;  // 0: lanes 0-15, 1: lanes 16-31
  int lm = lane & 15;    // M (for A/C) or N (for B) index

  int tileIdx = blockIdx.x * 4 + wave;  // 2048 tiles of 32x16 per channel
  int y0 = (tileIdx >> 6) << 5;
  int x0 = (tileIdx & 63) << 4;

  // ---- A: 13 weight chunks, branchless LDS reads, kept in registers ----
  v2f aReg[13];
#pragma unroll
  for (int k = 0; k < 13; ++k) {
    int i0 = 4 * k + 2 * half - lm + 15;  // always in [0, WTAB)
    aReg[k].x = wtab[i0];
    aReg[k].y = wtab[i0 + 1];
  }

  // ---- B: 17 input chunks (68 rows, replicate-clamped), preloaded ----
  v2f bReg[17];
#pragma unroll
  for (int k = 0; k < 17; ++k) {
    int p0 = 4 * k + 2 * half;
    int r0 = min(max(y0 - RAD + p0, 0), HH - 1);
    int r1 = min(max(y0 - RAD + p0 + 1, 0), HH - 1);
    bReg[k].x = in[r0 * WW + x0 + lm];
    bReg[k].y = in[r1 * WW + x0 + lm];
  }

  // ---- 26 back-to-back WMMAs, B registers shared between both tiles ----
  v8f c0 = {0.f, 0.f, 0.f, 0.f, 0.f, 0.f, 0.f, 0.f};
  v8f c1 = {0.f, 0.f, 0.f, 0.f, 0.f, 0.f, 0.f, 0.f};
#pragma unroll
  for (int k = 0; k < 13; ++k) {
    c0 = __builtin_amdgcn_wmma_f32_16x16x4_f32(false, aReg[k], false, bReg[k],
                                               (short)0, c0, false, false);
    c1 = __builtin_amdgcn_wmma_f32_16x16x4_f32(false, aReg[k], false,
                                               bReg[k + 4], (short)0, c1,
                                               false, false);
  }

  // ---- transpose both 16x16 result tiles via LDS, coalesced stores ----
  float* T = tile[wave];
#pragma unroll
  for (int r = 0; r < 8; ++r) T[lm * 17 + r + 8 * half] = c0[r];
  __syncthreads();
#pragma unroll
  for (int r = 0; r < 8; ++r) {
    int n = r + 8 * half;
    outT[(size_t)(x0 + n) * HH + y0 + lm] = T[n * 17 + lm];
  }
  __syncthreads();
#pragma unroll
  for (int r = 0; r < 8; ++r) T[lm * 17 + r + 8 * half] = c1[r];
  __syncthreads();
#pragma unroll
  for (int r = 0; r < 8; ++r) {
    int n = r + 8 * half;
    outT[(size_t)(x0 + n) * HH + y0 + 16 + lm] = T[n * 17 + lm];
  }
}

// ---------------------------------------------------------------------------
// update: div(v), semi-Lagrangian bilinear warp of z and img (index clamping
// matches map_coordinates(order=1, mode='nearest')), then the step updates.
// ---------------------------------------------------------------------------
__global__ __launch_bounds__(256) void update_k(const float* __restrict__ img,
                                                const float* __restrict__ z,
                                                const float* __restrict__ vx,
                                                const float* __restrict__ vy,
                                                float* __restrict__ imgN,
                                                float* __restrict__ zN) {
  int idx = blockIdx.x * 256 + threadIdx.x;
  int y = idx >> 10;
  int x = idx & (WW - 1);
  int xm = max(x - 1, 0), xp = min(x + 1, WW - 1);
  int ym = max(y - 1, 0), yp = min(y + 1, HH - 1);

  float dvxdx = 0.5f * (vx[y * WW + xp] - vx[y * WW + xm]);
  float dvydy = 0.5f * (vy[yp * WW + x] - vy[ym * WW + x]);
  float divv = dvxdx + dvydy;

  float cy = (float)y - vy[idx] * INV_N;
  float cx = (float)x - vx[idx] * INV_N;

  float fy0 = floorf(cy);
  float fx0 = floorf(cx);
  float ty = cy - fy0;
  float tx = cx - fx0;
  int iy0 = min(max((int)fy0, 0), HH - 1);
  int iy1 = min(max((int)fy0 + 1, 0), HH - 1);
  int ix0 = min(max((int)fx0, 0), WW - 1);
  int ix1 = min(max((int)fx0 + 1, 0), WW - 1);

  int o00 = iy0 * WW + ix0, o01 = iy0 * WW + ix1;
  int o10 = iy1 * WW + ix0, o11 = iy1 * WW + ix1;

  float zw = (z[o00] * (1.0f - tx) + z[o01] * tx) * (1.0f - ty) +
             (z[o10] * (1.0f - tx) + z[o11] * tx) * ty;
  float iw = (img[o00] * (1.0f - tx) + img[o01] * tx) * (1.0f - ty) +
             (img[o10] * (1.0f - tx) + img[o11] * tx) * ty;

  float znew = zw - z[idx] * divv * INV_N;
  zN[idx] = znew;
  imgN[idx] = iw + znew * INV_N;  // MU == 1
}

// ---------------------------------------------------------------------------
extern "C" void kernel_launch(void* const* d_in, const int* in_sizes, int n_in,
                              void* d_out, int out_size, void* d_ws,
                              size_t ws_size, hipStream_t stream) {
  (void)in_sizes;
  (void)n_in;
  (void)out_size;
  (void)ws_size;
  const float* img_in = (const float*)d_in[0];  // [1,1,1024,1024]
  const float* z_in = (const float*)d_in[1];    // [1024,1024]
  float* out = (float*)d_out;
  float* ws = (float*)d_ws;

  float* imgA = ws + (size_t)0 * NPIX;
  float* imgB = ws + (size_t)1 * NPIX;
  float* zA = ws + (size_t)2 * NPIX;
  float* zB = ws + (size_t)3 * NPIX;
  float* fx = ws + (size_t)4 * NPIX;  // force x, later reused as vx
  float* fy = ws + (size_t)5 * NPIX;  // force y, later reused as vy
  float* txT = ws + (size_t)6 * NPIX; // vertically blurred, transposed
  float* tyT = ws + (size_t)7 * NPIX;

  const float* curImg = img_in;
  const float* curZ = z_in;

  for (int s = 0; s < 16; ++s) {
    force_k<<<NPIX / 256, 256, 0, stream>>>(curImg, curZ, fx, fy);
    // vertical pass (writes transposed), then "vertical" pass on transposed
    // data = horizontal pass (writes back in original orientation).
    blurV_T<<<dim3(512, 2), 128, 0, stream>>>(fx, fy, txT, tyT);
    blurV_T<<<dim3(512, 2), 128, 0, stream>>>(txT, tyT, fx, fy);
    float* nImg = (s == 15) ? out : ((s & 1) ? imgB : imgA);
    float* nZ = (s & 1) ? zB : zA;
    update_k<<<NPIX / 256, 256, 0, stream>>>(curImg, curZ, fx, fy, nImg, nZ);
    curImg = nImg;
    curZ = nZ;
  }
}